// HyperConvD_75771813036395
// MI455X (gfx1250) — compile-verified
//
#include <hip/hip_runtime.h>
#include <math.h>

// ---- problem constants (match reference) ----
#define BATCH   512
#define DIM     200       // emb dim == R_NUM
#define DIMP    256       // padded N (4 x 64) for q/k/v/att
#define KQP     224       // padded K for D=200 GEMMs (mult of 32)
#define REL     1800
#define RELP    1824      // padded K for r GEMMs (mult of 32)
#define NCHAN   200
#define IMH     60
#define IMW     20
#define FH      58
#define FW      18
#define NPOS    (FH*FW)   // 1044
#define NPOSP   1056      // padded K for FC GEMM (mult of 32)
#define FCO     100
#define FCO_PAD 128       // padded N for FC GEMM (2 x 64)
#define EPS     1e-5f

typedef float v2f __attribute__((ext_vector_type(2)));
typedef float v8f __attribute__((ext_vector_type(8)));

struct Idx6 { const int* p[6]; };

// ------------------------------------------------------------------
__global__ void zero_range(float* __restrict__ p, int n) {
    for (int i = blockIdx.x * blockDim.x + threadIdx.x; i < n; i += gridDim.x * blockDim.x) p[i] = 0.f;
}

// bn0: per-slot stats over all B*D gathered embedding values
__global__ void bn0_stats(Idx6 idx, const float* __restrict__ E_w, float* __restrict__ slot_stats) {
    __shared__ float red[256], red2[256];
    int s = blockIdx.x, tid = threadIdx.x;
    const int* ip = idx.p[s];
    float s1 = 0.f, s2 = 0.f;
    for (int e = tid; e < BATCH * DIM; e += 256) {
        int b = e / DIM, d = e % DIM;
        float v = E_w[(size_t)ip[b] * DIM + d];
        s1 += v; s2 += v * v;
    }
    red[tid] = s1; red2[tid] = s2; __syncthreads();
    for (int off = 128; off > 0; off >>= 1) {
        if (tid < off) { red[tid] += red[tid + off]; red2[tid] += red2[tid + off]; }
        __syncthreads();
    }
    if (tid == 0) {
        float N = (float)(BATCH * DIM);
        float m = red[0] / N;
        float var = red2[0] / N - m * m;
        slot_stats[2 * s]     = m;
        slot_stats[2 * s + 1] = rsqrtf(var + EPS);
    }
}

// build normalized x (B,6*D) and e_mean (B, KQP zero-padded)
__global__ void build_x(Idx6 idx, const float* __restrict__ E_w, const float* __restrict__ slot_stats,
                        const float* __restrict__ g0, const float* __restrict__ b0,
                        float* __restrict__ x, float* __restrict__ e_mean) {
    int b = blockIdx.x, d = threadIdx.x;
    if (d >= DIM) return;
    float g = g0[0], be = b0[0];
    float acc = 0.f;
    for (int s = 0; s < 6; ++s) {
        float m = slot_stats[2 * s], inv = slot_stats[2 * s + 1];
        float v = (E_w[(size_t)idx.p[s][b] * DIM + d] - m) * inv * g + be;
        x[b * (6 * DIM) + s * DIM + d] = v;
        acc += v;
    }
    e_mean[b * KQP + d] = acc * (1.f / 6.f);
}

// gather r = R_w[r_idx] into zero-padded (B x RELP)
__global__ void gather_r(const int* __restrict__ ridx, const float* __restrict__ R_w, float* __restrict__ r) {
    int b = blockIdx.x, tid = threadIdx.x;
    const float* src = R_w + (size_t)ridx[b] * REL;
    for (int j = tid; j < REL; j += 256) r[b * RELP + j] = src[j];
}

// bn2: per-column norm over batch -> conv weights (B x REL)
__global__ void bn2_weights(const float* __restrict__ r, const float* __restrict__ g2,
                            const float* __restrict__ b2, float* __restrict__ w_conv) {
    __shared__ float red[128], red2[128];
    int j = blockIdx.x, tid = threadIdx.x;
    float s1 = 0.f, s2 = 0.f;
    for (int b = tid; b < BATCH; b += 128) { float v = r[b * RELP + j]; s1 += v; s2 += v * v; }
    red[tid] = s1; red2[tid] = s2; __syncthreads();
    for (int off = 64; off > 0; off >>= 1) {
        if (tid < off) { red[tid] += red[tid + off]; red2[tid] += red2[tid + off]; }
        __syncthreads();
    }
    float m = red[0] / (float)BATCH;
    float var = red2[0] / (float)BATCH - m * m;
    float sc = g2[j] * rsqrtf(var + EPS);
    float sh = b2[j] - m * sc;
    for (int b = tid; b < BATCH; b += 128) w_conv[b * REL + j] = r[b * RELP + j] * sc + sh;
}

// zero-padded copy: dst (rows_p x cols_p) <- src (rows x cols)
__global__ void pad_copy(const float* __restrict__ src, float* __restrict__ dst,
                         int rows, int cols, int rows_p, int cols_p) {
    int n = rows_p * cols_p;
    for (int i = blockIdx.x * blockDim.x + threadIdx.x; i < n; i += gridDim.x * blockDim.x) {
        int rr = i / cols_p, cc = i % cols_p;
        dst[i] = (rr < rows && cc < cols) ? src[rr * cols + cc] : 0.f;
    }
}

// transpose + zero-pad: src (K x N row-major) -> dst (Np x Kp row-major)
__global__ void transpose_pad(const float* __restrict__ src, float* __restrict__ dst,
                              int K, int N, int Kp, int Np) {
    int n = Np * Kp;
    for (int i = blockIdx.x * blockDim.x + threadIdx.x; i < n; i += gridDim.x * blockDim.x) {
        int rr = i / Kp, cc = i % Kp;                 // rr = N index, cc = K index
        dst[i] = (rr < N && cc < K) ? src[(size_t)cc * N + rr] : 0.f;
    }
}

// ------------------------------------------------------------------
// branchless f32 WMMA GEMM, B in N x K ("BT") layout:
//   C = alpha * A(MxK) * B^T,  M%16==0, KP%32==0, NP%64==0, lds even.
// Per wave: 16 rows x 64 cols (4 tiles). Loop body loads TWO 16-K
// sub-chunks (all contiguous b64 loads) then issues 2x16 WMMAs, so the
// second sub-chunk's loads stay in flight under the first 16 WMMAs.
// CT=true stores C transposed (contiguous per-lane stores).
// ------------------------------------------------------------------
template <bool CT>
__global__ void wmma_gemm_bt(const float* __restrict__ A, const float* __restrict__ B,
                             float* __restrict__ C, int M, int nGroups, int KP,
                             int lda, int ldb, int ldc, float alpha) {
    int wave = (int)(blockIdx.x * blockDim.x + threadIdx.x) >> 5;
    int lane = threadIdx.x & 31;
    int tm = wave / nGroups, tg = wave % nGroups;
    if (tm * 16 >= M) return;             // wave-uniform; EXEC all-1 inside
    int mbase = tm * 16, nbase = tg * 64;
    int lh  = lane >> 4;                  // lane half -> K pair
    int l16 = lane & 15;
    int lh2 = lh * 2;
    const float* __restrict__ Arow = A + (size_t)(mbase + l16) * lda;
    const float* Br[4];
    #pragma unroll
    for (int t = 0; t < 4; ++t) Br[t] = B + (size_t)(nbase + t * 16 + l16) * ldb;
    v8f acc[4] = {};
    for (int k0 = 0; k0 < KP; k0 += 32) {
        int kA = k0 + lh2;
        int kB = kA + 16;
        v2f aA[4], aB[4], bA[4][4], bB[4][4];
        #pragma unroll
        for (int s = 0; s < 4; ++s) {
            aA[s] = *(const v2f*)(Arow + kA + 4 * s);
            #pragma unroll
            for (int t = 0; t < 4; ++t) bA[s][t] = *(const v2f*)(Br[t] + kA + 4 * s);
        }
        #pragma unroll
        for (int s = 0; s < 4; ++s) {
            aB[s] = *(const v2f*)(Arow + kB + 4 * s);
            #pragma unroll
            for (int t = 0; t < 4; ++t) bB[s][t] = *(const v2f*)(Br[t] + kB + 4 * s);
        }
        #pragma unroll
        for (int s = 0; s < 4; ++s) {
            #pragma unroll
            for (int t = 0; t < 4; ++t)
                acc[t] = __builtin_amdgcn_wmma_f32_16x16x4_f32(false, aA[s], false, bA[s][t],
                                                               (short)0, acc[t], false, false);
        }
        #pragma unroll
        for (int s = 0; s < 4; ++s) {
            #pragma unroll
            for (int t = 0; t < 4; ++t)
                acc[t] = __builtin_amdgcn_wmma_f32_16x16x4_f32(false, aB[s], false, bB[s][t],
                                                               (short)0, acc[t], false, false);
        }
    }
    if (CT) {
        #pragma unroll
        for (int t = 0; t < 4; ++t) {
            float* Cc = C + (size_t)(nbase + t * 16 + l16) * ldc + mbase + 8 * lh;
            #pragma unroll
            for (int i = 0; i < 8; ++i) Cc[i] = alpha * acc[t][i];   // contiguous -> merged stores
        }
    } else {
        #pragma unroll
        for (int i = 0; i < 8; ++i) {
            float* Cr = C + (size_t)(mbase + i + 8 * lh) * ldc + nbase + l16;
            #pragma unroll
            for (int t = 0; t < 4; ++t) Cr[t * 16] = alpha * acc[t][i];
        }
    }
}

// row softmax on S (BATCH x BATCH), in place
__global__ void softmax_rows(float* __restrict__ S) {
    __shared__ float red[256];
    int row = blockIdx.x, tid = threadIdx.x;
    float* r = S + (size_t)row * BATCH;
    float m = -1e30f;
    for (int j = tid; j < BATCH; j += 256) m = fmaxf(m, r[j]);
    red[tid] = m; __syncthreads();
    for (int off = 128; off > 0; off >>= 1) {
        if (tid < off) red[tid] = fmaxf(red[tid], red[tid + off]);
        __syncthreads();
    }
    m = red[0]; __syncthreads();
    float s = 0.f;
    for (int j = tid; j < BATCH; j += 256) { float e = __expf(r[j] - m); r[j] = e; s += e; }
    red[tid] = s; __syncthreads();
    for (int off = 128; off > 0; off >>= 1) {
        if (tid < off) red[tid] += red[tid + off];
        __syncthreads();
    }
    float inv = 1.f / red[0];
    for (int j = tid; j < BATCH; j += 256) r[j] *= inv;
}

// ------------------------------------------------------------------
// conv pass 1: per-channel sum/sumsq of the grouped conv output
// ------------------------------------------------------------------
__global__ void conv_stats(const float* __restrict__ x, const float* __restrict__ w_conv,
                           float* __restrict__ chan_sum, float* __restrict__ chan_sq) {
    __shared__ float img[IMH * IMW];
    __shared__ float wt[REL];
    __shared__ float red[256], red2[256];
    int b = blockIdx.x, tid = threadIdx.x;
    for (int i = tid; i < IMH * IMW; i += 256) img[i] = x[b * (IMH * IMW) + i];
    for (int i = tid; i < REL; i += 256)       wt[i]  = w_conv[b * REL + i];
    __syncthreads();
    for (int c = 0; c < NCHAN; ++c) {
        const float* w9 = &wt[c * 9];
        float s1 = 0.f, s2 = 0.f;
        for (int pos = tid; pos < NPOS; pos += 256) {
            int h = pos / FW, w = pos % FW;
            float v = 0.f;
            #pragma unroll
            for (int i = 0; i < 3; ++i)
                #pragma unroll
                for (int j = 0; j < 3; ++j)
                    v = fmaf(img[(h + i) * IMW + (w + j)], w9[i * 3 + j], v);
            s1 += v; s2 += v * v;
        }
        red[tid] = s1; red2[tid] = s2; __syncthreads();
        for (int off = 128; off > 0; off >>= 1) {
            if (tid < off) { red[tid] += red[tid + off]; red2[tid] += red2[tid + off]; }
            __syncthreads();
        }
        if (tid == 0) { atomicAdd(&chan_sum[c], red[0]); atomicAdd(&chan_sq[c], red2[0]); }
        __syncthreads();
    }
}

__global__ void bn1_finalize(const float* __restrict__ chan_sum, const float* __restrict__ chan_sq,
                             const float* __restrict__ g, const float* __restrict__ be,
                             float* __restrict__ scale, float* __restrict__ shift) {
    int c = blockIdx.x * blockDim.x + threadIdx.x;
    if (c >= NCHAN) return;
    const float N = (float)BATCH * (float)NPOS;
    float m = chan_sum[c] / N;
    float var = chan_sq[c] / N - m * m;
    float s = g[c] * rsqrtf(var + EPS);
    scale[c] = s;
    shift[c] = be[c] - m * s;
}

// conv pass 2: recompute conv, bn1 + relu + attention-weighted channel sum
__global__ void conv_apply(const float* __restrict__ x, const float* __restrict__ w_conv,
                           const float* __restrict__ att, const float* __restrict__ scale,
                           const float* __restrict__ shift, float* __restrict__ y2) {
    __shared__ float img[IMH * IMW];
    __shared__ float wt[REL];
    __shared__ float arow[NCHAN], sc[NCHAN], sh[NCHAN];
    int b = blockIdx.x, tid = threadIdx.x;
    for (int i = tid; i < IMH * IMW; i += 256) img[i] = x[b * (IMH * IMW) + i];
    for (int i = tid; i < REL; i += 256)       wt[i]  = w_conv[b * REL + i];
    for (int i = tid; i < NCHAN; i += 256) { arow[i] = att[b * DIMP + i]; sc[i] = scale[i]; sh[i] = shift[i]; }
    __syncthreads();
    float acc[5] = {0.f, 0.f, 0.f, 0.f, 0.f};
    for (int c = 0; c < NCHAN; ++c) {
        const float* w9 = &wt[c * 9];
        float a = arow[c], s = sc[c], t = sh[c];
        int pi = 0;
        for (int pos = tid; pos < NPOS; pos += 256, ++pi) {
            int h = pos / FW, w = pos % FW;
            float v = 0.f;
            #pragma unroll
            for (int i = 0; i < 3; ++i)
                #pragma unroll
                for (int j = 0; j < 3; ++j)
                    v = fmaf(img[(h + i) * IMW + (w + j)], w9[i * 3 + j], v);
            v = fmaf(v, s, t);
            v = v > 0.f ? v : 0.f;
            acc[pi] = fmaf(a, v, acc[pi]);
        }
    }
    int pi = 0;
    for (int pos = tid; pos < NPOS; pos += 256, ++pi) y2[b * NPOSP + pos] = acc[pi];
}

// scores[b] = sum_o (fc_out[b,o] + bias[o]) * p[o]
__global__ void fc_final(const float* __restrict__ fc_out, const float* __restrict__ bias,
                         const float* __restrict__ p, float* __restrict__ out) {
    int b = blockIdx.x * blockDim.x + threadIdx.x;
    if (b >= BATCH) return;
    float s = 0.f;
    for (int o = 0; o < FCO; ++o) s = fmaf(fc_out[b * FCO_PAD + o] + bias[o], p[o], s);
    out[b] = s;
}

// ------------------------------------------------------------------
extern "C" void kernel_launch(void* const* d_in, const int* in_sizes, int n_in,
                              void* d_out, int out_size, void* d_ws, size_t ws_size,
                              hipStream_t stream) {
    const int*   r_idx = (const int*)d_in[0];
    Idx6 idx;
    for (int i = 0; i < 6; ++i) idx.p[i] = (const int*)d_in[1 + i];
    const float* E_w   = (const float*)d_in[7];
    const float* R_w   = (const float*)d_in[8];
    const float* bn0_g = (const float*)d_in[9];
    const float* bn0_b = (const float*)d_in[10];
    const float* bn1_g = (const float*)d_in[11];
    const float* bn1_b = (const float*)d_in[12];
    const float* bn2_g = (const float*)d_in[13];
    const float* bn2_b = (const float*)d_in[14];
    const float* Qm    = (const float*)d_in[15];
    const float* Km    = (const float*)d_in[16];
    const float* Vm    = (const float*)d_in[17];
    const float* fc6_W = (const float*)d_in[18];
    const float* fc6_b = (const float*)d_in[19];
    const float* pvec  = (const float*)d_in[20];
    float* out = (float*)d_out;

    // workspace carve-out (floats); every size even -> 8B alignment holds
    float* ws = (float*)d_ws;
    size_t off = 0;
    float* x        = ws + off; off += (size_t)BATCH * 6 * DIM;
    float* e_mean   = ws + off; off += (size_t)BATCH * KQP;      // 512 x 224 (zero-padded)
    float* rP       = ws + off; off += (size_t)BATCH * RELP;     // 512 x 1824 (zero-padded)
    float* w_conv   = ws + off; off += (size_t)BATCH * REL;
    float* qb       = ws + off; off += (size_t)BATCH * DIMP;     // 512 x 256
    float* kb       = ws + off; off += (size_t)BATCH * DIMP;     // 512 x 256 (N x K for S gemm)
    float* vbT      = ws + off; off += (size_t)DIMP * BATCH;     // 256 x 512 (N x K for att gemm)
    float* S        = ws + off; off += (size_t)BATCH * BATCH;
    float* attb     = ws + off; off += (size_t)BATCH * DIMP;
    float* y2       = ws + off; off += (size_t)BATCH * NPOSP;    // 512 x 1056 (zero-padded)
    float* fc_out   = ws + off; off += (size_t)BATCH * FCO_PAD;
    float* Qt       = ws + off; off += (size_t)DIMP * KQP;       // 256 x 224
    float* Kt       = ws + off; off += (size_t)DIMP * RELP;      // 256 x 1824
    float* Vt       = ws + off; off += (size_t)DIMP * RELP;      // 256 x 1824
    float* Wp       = ws + off; off += (size_t)FCO_PAD * NPOSP;  // 128 x 1056
    float* slot_st  = ws + off; off += 16;
    float* chan_sum = ws + off; off += NCHAN;
    float* chan_sq  = ws + off; off += NCHAN;
    float* bn1_sc   = ws + off; off += NCHAN;
    float* bn1_sh   = ws + off; off += NCHAN;
    (void)ws_size; (void)in_sizes; (void)n_in; (void)out_size;

    auto gemm = [&](bool ct, const float* A, const float* Bm, float* C,
                    int M, int NP, int KP, int lda, int ldb, int ldc, float alpha) {
        int nGroups = NP / 64;
        int waves   = (M / 16) * nGroups;
        int grid    = (waves * 32 + 255) / 256;
        if (ct) wmma_gemm_bt<true ><<<grid, 256, 0, stream>>>(A, Bm, C, M, nGroups, KP, lda, ldb, ldc, alpha);
        else    wmma_gemm_bt<false><<<grid, 256, 0, stream>>>(A, Bm, C, M, nGroups, KP, lda, ldb, ldc, alpha);
    };

    // 0. zero-fill padded buffers, build transposed/padded operands
    zero_range<<<64, 256, 0, stream>>>(e_mean, BATCH * KQP);
    zero_range<<<256, 256, 0, stream>>>(rP, BATCH * RELP);
    zero_range<<<256, 256, 0, stream>>>(y2, BATCH * NPOSP);
    zero_range<<<1, 256, 0, stream>>>(chan_sum, NCHAN);
    zero_range<<<1, 256, 0, stream>>>(chan_sq, NCHAN);
    transpose_pad<<<64, 256, 0, stream>>>(Qm, Qt, DIM, DIM, KQP, DIMP);     // 256 x 224
    transpose_pad<<<256, 256, 0, stream>>>(Km, Kt, REL, DIM, RELP, DIMP);   // 256 x 1824
    transpose_pad<<<256, 256, 0, stream>>>(Vm, Vt, REL, DIM, RELP, DIMP);   // 256 x 1824
    pad_copy<<<128, 256, 0, stream>>>(fc6_W, Wp, FCO, NPOS, FCO_PAD, NPOSP);// 128 x 1056

    // 1. bn0 stats + normalized x / e_mean
    bn0_stats<<<6, 256, 0, stream>>>(idx, E_w, slot_st);
    build_x<<<BATCH, 256, 0, stream>>>(idx, E_w, slot_st, bn0_g, bn0_b, x, e_mean);

    // 2. relation gather + bn2 -> per-sample conv weights
    gather_r<<<BATCH, 256, 0, stream>>>(r_idx, R_w, rP);
    bn2_weights<<<REL, 128, 0, stream>>>(rP, bn2_g, bn2_b, w_conv);

    // 3. attention GEMMs (all BT; b64 fragment loads; 32 WMMAs per wait window)
    gemm(false, e_mean, Qt, qb,  BATCH, DIMP,  KQP,  KQP,  KQP,  DIMP,  1.f);  // q = e_mean @ Q
    gemm(false, rP,     Kt, kb,  BATCH, DIMP,  RELP, RELP, RELP, DIMP,  1.f);  // k = r @ K
    gemm(true,  rP,     Vt, vbT, BATCH, DIMP,  RELP, RELP, RELP, BATCH, 1.f);  // v^T = (r @ V)^T
    gemm(false, qb,     kb, S,   BATCH, BATCH, KQP,  DIMP, DIMP, BATCH,
         1.f / sqrtf((float)DIM));                                             // S = q @ k^T / sqrt(D)
    softmax_rows<<<BATCH, 256, 0, stream>>>(S);
    gemm(false, S, vbT, attb, BATCH, DIMP, BATCH, BATCH, BATCH, DIMP, 1.f);    // att = softmax(S) @ v

    // 4. grouped conv: stats pass -> bn1 -> apply pass (relu + weighted channel sum)
    conv_stats<<<BATCH, 256, 0, stream>>>(x, w_conv, chan_sum, chan_sq);
    bn1_finalize<<<1, 256, 0, stream>>>(chan_sum, chan_sq, bn1_g, bn1_b, bn1_sc, bn1_sh);
    conv_apply<<<BATCH, 256, 0, stream>>>(x, w_conv, attb, bn1_sc, bn1_sh, y2);

    // 5. FC (WMMA) + final projection
    gemm(false, y2, Wp, fc_out, BATCH, FCO_PAD, NPOSP, NPOSP, NPOSP, FCO_PAD, 1.f);
    fc_final<<<2, 256, 0, stream>>>(fc_out, fc6_b, pvec, out);
}